// Encoder_38697655337457
// MI455X (gfx1250) — compile-verified
//
#include <hip/hip_runtime.h>
#include <hip/hip_bf16.h>
#include <math.h>

// Problem constants (from reference): B=128, S=512, H=1024, L=4, V=32000
#define BB   128
#define SS   512
#define HH   1024
#define LL   4
#define G4   4096   // 4*H gate width
#define NKC  4      // split-K chunks
#define KC   (HH / NKC)   // 256 K columns per chunk

typedef __attribute__((ext_vector_type(2))) float v2f;
typedef __attribute__((ext_vector_type(4))) float v4f;
typedef __attribute__((ext_vector_type(8))) float v8f;

// ---------------------------------------------------------------------------
// 1) lengths[b] = sum_s (x[b,s] > 0)
// ---------------------------------------------------------------------------
__global__ __launch_bounds__(256) void lengths_kernel(const int* __restrict__ x,
                                                      int* __restrict__ lengths) {
    __shared__ int sdata[256];
    const int b   = blockIdx.x;
    const int tid = threadIdx.x;
    const int* row = x + b * SS;
    int c = (row[tid] > 0) + (row[tid + 256] > 0);
    sdata[tid] = c;
    __syncthreads();
    for (int s = 128; s > 0; s >>= 1) {
        if (tid < s) sdata[tid] += sdata[tid + s];
        __syncthreads();
    }
    if (tid == 0) lengths[b] = sdata[0];
}

// ---------------------------------------------------------------------------
// 2) Stable descending argsort of 128 lengths via rank counting.
//    rank[b] == unsort_idx[b]; sortedIdx[r] = original row at sorted pos r;
//    bsLast = #rows with length >= S.
// ---------------------------------------------------------------------------
__global__ __launch_bounds__(BB) void sort_kernel(const int* __restrict__ lengths,
                                                  int* __restrict__ rank,
                                                  int* __restrict__ sortedIdx,
                                                  int* __restrict__ bsLast) {
    __shared__ int len[BB];
    const int b = threadIdx.x;
    len[b] = lengths[b];
    __syncthreads();
    const int lb = len[b];
    int r = 0;
    for (int j = 0; j < BB; ++j) {
        const int lj = len[j];
        if (lj > lb || (lj == lb && j < b)) ++r;   // stable, descending
    }
    rank[b] = r;
    sortedIdx[r] = b;
    if (b == 0) {
        int c = 0;
        for (int j = 0; j < BB; ++j) c += (len[j] >= SS) ? 1 : 0;
        *bsLast = c;
    }
}

// ---------------------------------------------------------------------------
// 3) x0[bs, :] = emb[ x[sortedIdx[bs], 0], : ]   (t = 0 tokens, sorted order)
// ---------------------------------------------------------------------------
__global__ __launch_bounds__(256) void embed_kernel(const int* __restrict__ x,
                                                    const float* __restrict__ emb,
                                                    const int* __restrict__ sortedIdx,
                                                    float* __restrict__ x0) {
    const int bs  = blockIdx.x;
    const int tok = x[sortedIdx[bs] * SS];      // token at t=0
    const v4f* src = (const v4f*)(emb + (size_t)tok * HH);
    v4f*       dst = (v4f*)(x0 + (size_t)bs * HH);
    dst[threadIdx.x] = src[threadIdx.x];        // 256 threads * 4 f32 = 1024 f32
}

// ---------------------------------------------------------------------------
// 4a) Split-K WMMA GEMM for the i/g/o gate blocks (f gate & W_hh eliminated).
//     One wave computes a 16x16 (M x N) tile of all 3 gates over one 256-wide
//     K chunk using V_WMMA_F32_16X16X4_F32, with a register double-buffered
//     pipeline: loads for stage s+1 are issued BEFORE the WMMAs of stage s,
//     so each wait covers a full stage of load latency. Weight stream loads
//     are non-temporal (each element read exactly once).
//
//     partial layout: [NKC][3][128][1024]  (gate 0=i, 1=g, 2=o)
//
//     Fragment layouts (ISA 7.12.2, 32-bit, wave32):
//       A 16x4 : row = m0 + (lane&15); K pair {k,k+1} (lanes 0-15) or
//                {k+2,k+3} (lanes 16-31)
//       B 4x16 : col = n0 + (lane&15); same K split per half-wave
//       C/D    : elem e of v8f -> M = m0 + e + 8*(lane>=16), N = n0 + (lane&15)
// ---------------------------------------------------------------------------
#define KSTAGE   16                // K columns per pipeline stage (4 wmma steps)
#define NSTAGES  (KC / KSTAGE)     // 16 stages per chunk

__device__ __forceinline__ void load_stage(const float* arow, const float* wi,
                                           const float* wg, const float* wo,
                                           int k, v2f* a, v2f* bi, v2f* bg, v2f* bo) {
#pragma unroll
    for (int q = 0; q < 4; ++q) {
        a[q]  = *(const v2f*)(arow + k + 4 * q);
        bi[q] = __builtin_nontemporal_load((const v2f*)(wi + k + 4 * q));
        bg[q] = __builtin_nontemporal_load((const v2f*)(wg + k + 4 * q));
        bo[q] = __builtin_nontemporal_load((const v2f*)(wo + k + 4 * q));
    }
}

__global__ __launch_bounds__(32) void lstm_gemm_wmma(
    const float* __restrict__ xin,     // [128,1024] layer input (sorted order)
    const float* __restrict__ W,       // [4096,1024] W_ih for this layer
    float* __restrict__ partial)       // [NKC][3][128][1024]
{
    const int m0     = blockIdx.x * 16;   // 8 tiles over batch (M = 128)
    const int n0     = blockIdx.y * 16;   // 64 tiles over hidden (N = 1024)
    const int kc     = blockIdx.z;        // 4 K chunks of 256
    const int lane   = threadIdx.x;
    const int lane15 = lane & 15;
    const int khalf  = (lane >> 4) * 2;   // K sub-offset: 0 (lanes 0-15), 2 (16-31)
    const int kBase  = kc * KC;

    v8f acc_i = {}; v8f acc_g = {}; v8f acc_o = {};

    const float* arow = xin + (size_t)(m0 + lane15) * HH + khalf + kBase;
    const float* wi   = W + (size_t)(0 * HH + n0 + lane15) * HH + khalf + kBase;
    const float* wg   = W + (size_t)(2 * HH + n0 + lane15) * HH + khalf + kBase;
    const float* wo   = W + (size_t)(3 * HH + n0 + lane15) * HH + khalf + kBase;

    // Register double buffers: 2 stages x 4 k-steps x {a, bi, bg, bo}
    v2f A[2][4], BI[2][4], BG[2][4], BO[2][4];

    load_stage(arow, wi, wg, wo, 0, A[0], BI[0], BG[0], BO[0]);

#pragma unroll
    for (int s = 0; s < NSTAGES; ++s) {
        const int cur = s & 1;
        const int nxt = cur ^ 1;
        if (s + 1 < NSTAGES) {
            // issue next stage's 16 loads before consuming current stage
            load_stage(arow, wi, wg, wo, (s + 1) * KSTAGE,
                       A[nxt], BI[nxt], BG[nxt], BO[nxt]);
        }
#pragma unroll
        for (int q = 0; q < 4; ++q) {
            acc_i = __builtin_amdgcn_wmma_f32_16x16x4_f32(false, A[cur][q], false, BI[cur][q],
                                                          (short)0, acc_i, false, false);
            acc_g = __builtin_amdgcn_wmma_f32_16x16x4_f32(false, A[cur][q], false, BG[cur][q],
                                                          (short)0, acc_g, false, false);
            acc_o = __builtin_amdgcn_wmma_f32_16x16x4_f32(false, A[cur][q], false, BO[cur][q],
                                                          (short)0, acc_o, false, false);
        }
    }

    // Store raw partial gate sums (bias/activation applied in combine kernel).
    const int n     = n0 + lane15;
    const int mbase = m0 + ((lane >> 4) << 3);
    float* pi = partial + ((size_t)(kc * 3 + 0) * BB) * HH;
    float* pg = partial + ((size_t)(kc * 3 + 1) * BB) * HH;
    float* po = partial + ((size_t)(kc * 3 + 2) * BB) * HH;

#pragma unroll
    for (int e = 0; e < 8; ++e) {
        const size_t ro = (size_t)(mbase + e) * HH + n;
        pi[ro] = acc_i[e];
        pg[ro] = acc_g[e];
        po[ro] = acc_o[e];
    }
}

// ---------------------------------------------------------------------------
// 4b) Combine split-K partials (fixed order -> deterministic), add b_ih+b_hh,
//     apply LSTM activations:  c = sigmoid(i)*tanh(g); h = sigmoid(o)*tanh(c)
// ---------------------------------------------------------------------------
__global__ __launch_bounds__(256) void lstm_combine(
    const float* __restrict__ partial,   // [NKC][3][128][1024]
    const float* __restrict__ bih,       // [4096]
    const float* __restrict__ bhh,       // [4096]
    float* __restrict__ hout,            // [128,1024]
    float* __restrict__ cout)            // [128,1024]
{
    const int m = blockIdx.x;                       // 128
    const int n = blockIdx.y * 256 + threadIdx.x;   // 4 * 256 = 1024
    const size_t ro = (size_t)m * HH + n;

    float gi = 0.f, gg = 0.f, go = 0.f;
#pragma unroll
    for (int kc = 0; kc < NKC; ++kc) {
        gi += partial[((size_t)(kc * 3 + 0) * BB) * HH + ro];
        gg += partial[((size_t)(kc * 3 + 1) * BB) * HH + ro];
        go += partial[((size_t)(kc * 3 + 2) * BB) * HH + ro];
    }
    gi += bih[n]          + bhh[n];
    gg += bih[2 * HH + n] + bhh[2 * HH + n];
    go += bih[3 * HH + n] + bhh[3 * HH + n];

    const float si = 1.0f / (1.0f + expf(-gi));
    const float so = 1.0f / (1.0f + expf(-go));
    const float c  = si * tanhf(gg);
    const float h  = so * tanhf(c);
    cout[ro] = c;
    hout[ro] = h;
}

// ---------------------------------------------------------------------------
// 5) output[t, b, :] = (t < lengths[b]) ? h_top[rank[b], :] : 0
//    256 MB write-once stream -> non-temporal stores (don't thrash 192MB L2);
//    the 512 KB source row set stays L2-resident on the read path.
// ---------------------------------------------------------------------------
__global__ __launch_bounds__(256) void output_kernel(const float* __restrict__ hs3,
                                                     const int* __restrict__ lengths,
                                                     const int* __restrict__ rank,
                                                     float* __restrict__ out) {
    const int b   = blockIdx.x;     // 128
    const int t   = blockIdx.y;     // 512
    const int col = threadIdx.x * 4;
    v4f v = (v4f)(0.0f);
    if (t < lengths[b]) {
        v = *(const v4f*)(hs3 + (size_t)rank[b] * HH + col);
    }
    __builtin_nontemporal_store(v, (v4f*)(out + (((size_t)t * BB + b) * HH + col)));
}

// ---------------------------------------------------------------------------
// 6) state_h/state_c [L, B, H] in SORTED order, rows >= bsLast zeroed.
// ---------------------------------------------------------------------------
__global__ __launch_bounds__(256) void state_kernel(const float* __restrict__ hs,
                                                    const float* __restrict__ cs,
                                                    const int* __restrict__ bsLast,
                                                    float* __restrict__ outh,
                                                    float* __restrict__ outc) {
    const int bs  = blockIdx.x;     // 128
    const int l   = blockIdx.y;     // 4
    const int col = threadIdx.x * 4;
    const size_t off = ((size_t)l * BB + bs) * HH + col;
    v4f h = (v4f)(0.0f);
    v4f c = (v4f)(0.0f);
    if (bs < *bsLast) {
        h = *(const v4f*)(hs + off);
        c = *(const v4f*)(cs + off);
    }
    *(v4f*)(outh + off) = h;
    *(v4f*)(outc + off) = c;
}

// ---------------------------------------------------------------------------
// Launcher. Inputs: 0=x(int[128,512]) 1=emb(f32[32000,1024])
//                   2=W_ih(f32[4,4096,1024]) 3=W_hh(unused: h0==0)
//                   4=b_ih(f32[4,4096]) 5=b_hh(f32[4,4096])
// Output: concat(output[512,128,1024], state_h[4,128,1024], state_c[4,128,1024])
// ---------------------------------------------------------------------------
extern "C" void kernel_launch(void* const* d_in, const int* in_sizes, int n_in,
                              void* d_out, int out_size, void* d_ws, size_t ws_size,
                              hipStream_t stream) {
    const int*   x    = (const int*)d_in[0];
    const float* emb  = (const float*)d_in[1];
    const float* W_ih = (const float*)d_in[2];
    const float* b_ih = (const float*)d_in[4];
    const float* b_hh = (const float*)d_in[5];
    float*       out  = (float*)d_out;

    // Workspace layout (~11 MB): x0 | hs[L] | cs[L] | splitK partials | ints
    float* ws_f    = (float*)d_ws;
    float* x0      = ws_f;                               // 128*1024
    float* hs      = x0 + (size_t)BB * HH;               // 4*128*1024
    float* cs      = hs + (size_t)LL * BB * HH;          // 4*128*1024
    float* partial = cs + (size_t)LL * BB * HH;          // 4*3*128*1024
    int*   ws_i      = (int*)(partial + (size_t)NKC * 3 * BB * HH);
    int*   lengths   = ws_i;                             // 128
    int*   rank      = ws_i + BB;                        // 128 (== unsort_idx)
    int*   sortedIdx = ws_i + 2 * BB;                    // 128
    int*   bsLast    = ws_i + 3 * BB;                    // 1

    lengths_kernel<<<BB, 256, 0, stream>>>(x, lengths);
    sort_kernel<<<1, BB, 0, stream>>>(lengths, rank, sortedIdx, bsLast);
    embed_kernel<<<BB, 256, 0, stream>>>(x, emb, sortedIdx, x0);

    for (int l = 0; l < LL; ++l) {
        const float* xin = (l == 0) ? x0 : hs + (size_t)(l - 1) * BB * HH;
        lstm_gemm_wmma<<<dim3(BB / 16, HH / 16, NKC), 32, 0, stream>>>(
            xin, W_ih + (size_t)l * G4 * HH, partial);
        lstm_combine<<<dim3(BB, HH / 256), 256, 0, stream>>>(
            partial,
            b_ih + (size_t)l * G4,
            b_hh + (size_t)l * G4,
            hs + (size_t)l * BB * HH,
            cs + (size_t)l * BB * HH);
    }

    output_kernel<<<dim3(BB, SS), 256, 0, stream>>>(hs + (size_t)3 * BB * HH,
                                                    lengths, rank, out);

    float* outh = out + (size_t)SS * BB * HH;
    float* outc = outh + (size_t)LL * BB * HH;
    state_kernel<<<dim3(BB, LL), 256, 0, stream>>>(hs, cs, bsLast, outh, outc);
}